// RealFullAttention_37417755083570
// MI455X (gfx1250) — compile-verified
//
#include <hip/hip_runtime.h>
#include <hip/hip_bf16.h>

// ---------------------------------------------------------------------------
// GQA decode attention for MI455X (gfx1250, wave32, WMMA).
// HID=2048, H=16 q-heads, KV=2 kv-heads, D=256, REP=8, CACHE=32768.
// Outputs (flat): y[2048] | k_full[2*32769*256] | v_full[2*32769*256]
// Memory-bound (~480MB traffic, ~0.58 GFLOP) -> keep fp32; QK^T and PV on
// the matrix pipe via V_WMMA_F32_16X16X4_F32. One-shot streams (weights,
// old caches) use non-temporal loads so the appended K/V stays hot in L2.
// ---------------------------------------------------------------------------

typedef float v2f __attribute__((ext_vector_type(2)));
typedef float v4f __attribute__((ext_vector_type(4)));
typedef float v8f __attribute__((ext_vector_type(8)));

#define HID      2048
#define NHQ      16
#define NKV      2
#define HD       256
#define REPQ     8
#define SCACHE   32768
#define STOT     32769           // cache + 1 new token
#define CHUNK    2048
#define NCHUNK   17              // ceil(32769/2048)
#define NWAVE    8
#define NPART_G  (NCHUNK * NWAVE)         // 136 partials per kv-group
#define PART_STRIDE (8 * (2 + HD))        // 8 rows * (m,l + 256 f32) = 2064
// workspace float offsets
#define WS_QKV   0               // q(scaled)[4096] k_new[512] v_new[512]
#define WS_ATTN  5120            // attention result [4096]
#define WS_PART  9216            // partials: 2*136*2064 floats (~2.3MB)

__device__ __forceinline__ v8f wmma_f32_4(v2f a, v2f b, v8f c) {
  return __builtin_amdgcn_wmma_f32_16x16x4_f32(
      /*neg_a=*/false, a, /*neg_b=*/false, b,
      /*c_mod=*/(short)0, c, /*reuse_a=*/false, /*reuse_b=*/false);
}

// ---- DPP 16-lane (one DPP row) butterfly reductions: pure VALU, no LDS ----
template <int CTRL>
__device__ __forceinline__ float dpp_mov_f32(float v) {
  return __builtin_bit_cast(
      float, __builtin_amdgcn_update_dpp(0, __builtin_bit_cast(int, v),
                                         CTRL, 0xF, 0xF, true));
}
__device__ __forceinline__ float row16_max(float v) {
  v = fmaxf(v, dpp_mov_f32<0xB1>(v));   // quad_perm [1,0,3,2]  : xor 1
  v = fmaxf(v, dpp_mov_f32<0x4E>(v));   // quad_perm [2,3,0,1]  : xor 2
  v = fmaxf(v, dpp_mov_f32<0x141>(v));  // row_half_mirror      : xor 4 (eff.)
  v = fmaxf(v, dpp_mov_f32<0x140>(v));  // row_mirror           : xor 8 (eff.)
  return v;
}
__device__ __forceinline__ float row16_sum(float v) {
  v += dpp_mov_f32<0xB1>(v);
  v += dpp_mov_f32<0x4E>(v);
  v += dpp_mov_f32<0x141>(v);
  v += dpp_mov_f32<0x140>(v);
  return v;
}

// ---------------------------------------------------------------------------
// Kernel 1: fused QKV projection. 20 blocks x 256 threads, one column each.
// Softmax scale (D^-1/2 = 0.0625) baked into q. Weights streamed NT.
// ---------------------------------------------------------------------------
__global__ void qkv_proj_kernel(const float* __restrict__ x,
                                const float* __restrict__ Wq,
                                const float* __restrict__ Wk,
                                const float* __restrict__ Wv,
                                float* __restrict__ ws) {
  __shared__ float xs[HID];
  for (int i = threadIdx.x; i < HID; i += 256) xs[i] = x[i];
  __syncthreads();

  int c = blockIdx.x * 256 + threadIdx.x;
  const float* W;
  int col, ncol;
  float mul;
  if (c < 4096)      { W = Wq; col = c;        ncol = 4096; mul = 0.0625f; }
  else if (c < 4608) { W = Wk; col = c - 4096; ncol = 512;  mul = 1.0f;    }
  else               { W = Wv; col = c - 4608; ncol = 512;  mul = 1.0f;    }

  float acc = 0.0f;
#pragma unroll 8
  for (int i = 0; i < HID; ++i)
    acc += xs[i] * __builtin_nontemporal_load(&W[(size_t)i * ncol + col]);
  ws[WS_QKV + c] = acc * mul;
}

// ---------------------------------------------------------------------------
// Kernel 2: copy caches into output (appended layout) + write new token row.
// Sources are read-once -> NT loads; destinations stay RT (re-read by attn,
// ~134MB which nearly fits the 192MB L2). v4f = clang ext-vector so the
// nontemporal builtin accepts it; still lowers to b128 loads/stores.
// ---------------------------------------------------------------------------
__global__ void cache_copy_kernel(const float* __restrict__ kc,
                                  const float* __restrict__ vc,
                                  const float* __restrict__ kvnew, // k_new[512] v_new[512]
                                  float* __restrict__ kout,
                                  float* __restrict__ vout) {
  const long long ROW4    = HD / 4;                    // 64 v4f per row
  const long long PER     = (long long)NKV * SCACHE * ROW4;
  const long long GSTRIDE = (long long)gridDim.x * blockDim.x;
  long long gid = (long long)blockIdx.x * blockDim.x + threadIdx.x;

  const v4f* kc4 = (const v4f*)kc;
  const v4f* vc4 = (const v4f*)vc;
  v4f* ko4 = (v4f*)kout;
  v4f* vo4 = (v4f*)vout;

  for (long long t = gid; t < PER; t += GSTRIDE) {
    long long g   = t / ((long long)SCACHE * ROW4);
    long long rem = t - g * ((long long)SCACHE * ROW4);
    long long dst = g * ((long long)STOT * ROW4) + rem;
    ko4[dst] = __builtin_nontemporal_load(&kc4[t]);
    vo4[dst] = __builtin_nontemporal_load(&vc4[t]);
  }
  // new token rows: 2 tensors * 2 groups * 64 v4f
  if (gid < 256) {
    int tensor = (int)(gid >> 7);
    int g      = (int)((gid >> 6) & 1);
    int d4     = (int)(gid & 63);
    const v4f* src = (const v4f*)(kvnew + (tensor ? 512 : 0) + g * HD) + d4;
    v4f* dstb = tensor ? vo4 : ko4;
    dstb[(long long)g * STOT * ROW4 + (long long)SCACHE * ROW4 + d4] = *src;
  }
}

// ---------------------------------------------------------------------------
// Kernel 3: flash-style partial attention with fp32 WMMA.
// grid = NKV*NCHUNK blocks; 8 waves/block, each wave owns 16-position tiles.
// __launch_bounds__(256,1): few blocks in flight -> trade occupancy for
// VGPRs so the 16x v8f output accumulator never spills to scratch.
// ---------------------------------------------------------------------------
#define QS_STRIDE 260   // 16 rows, padded: banks (4*m + k) % 64 conflict-free
#define PS_STRIDE 18    // P scratch row stride, conflict-free + 8B alignable

__global__ void __launch_bounds__(256, 1)
attn_partial_kernel(const float* __restrict__ ws,     // q at WS_QKV
                    const float* __restrict__ kfull,  // [NKV][STOT][HD]
                    const float* __restrict__ vfull,
                    float* __restrict__ partials) {
  __shared__ float qs[16 * QS_STRIDE];
  __shared__ float ps[NWAVE][16 * PS_STRIDE];

  const int g     = blockIdx.x / NCHUNK;
  const int chunk = blockIdx.x % NCHUNK;
  const int tid   = threadIdx.x;

  // stage q (zero-padded to 16 rows) into LDS
  for (int i = tid; i < 16 * HD; i += 256) {
    int r = i / HD, d = i - r * HD;
    qs[r * QS_STRIDE + d] = (r < REPQ) ? ws[WS_QKV + (g * REPQ + r) * HD + d] : 0.0f;
  }
  __syncthreads();

  const int wave  = tid >> 5;
  const int lane  = tid & 31;
  const int n     = lane & 15;       // A row index == B column index
  const int khalf = lane >> 4;       // 0: K={0,1}, 1: K={2,3}

  const float* Kbase = kfull + (size_t)g * STOT * HD;
  const float* Vbase = vfull + (size_t)g * STOT * HD;
  float* myps = &ps[wave][0];

  const int jstart = chunk * CHUNK;
  const int jend   = (jstart + CHUNK < STOT) ? (jstart + CHUNK) : STOT;

  float m_run[8], l_run[8];
  v8f o_acc[16];
#pragma unroll
  for (int i = 0; i < 8; ++i) { m_run[i] = -INFINITY; l_run[i] = 0.0f; }
#pragma unroll
  for (int dc = 0; dc < 16; ++dc) o_acc[dc] = (v8f){};

  for (int jt = jstart + wave * 16; jt < jend; jt += NWAVE * 16) {
    // prefetch next tile's K/V rows for this lane (global_prefetch_b8)
    {
      const int jnext = jt + NWAVE * 16 + n;
      if (jnext < jend) {
        __builtin_prefetch(Kbase + (size_t)jnext * HD + khalf * 128, 0, 3);
        __builtin_prefetch(Vbase + (size_t)jnext * HD + khalf * 128, 0, 3);
      }
    }

    // ---- scores: S[16x16] = Q(16x256) * K^T(256x16), 64 WMMA k-steps ----
    v8f s = (v8f){};
    const float* qrow = qs + (size_t)n * QS_STRIDE;
    int jb = jt + n; if (jb > STOT - 1) jb = STOT - 1;
    const float* krow = Kbase + (size_t)jb * HD;
#pragma unroll 8
    for (int kk = 0; kk < 64; ++kk) {
      const int k0 = kk * 4 + khalf * 2;
      v2f a = *(const v2f*)(qrow + k0);
      v2f b = *(const v2f*)(krow + k0);
      s = wmma_f32_4(a, b, s);
    }

    // ---- mask tail columns, per-row online softmax over 16-lane halves ----
    const bool colvalid = (jt + n) < jend;
    float sc[8];
#pragma unroll
    for (int i = 0; i < 8; ++i) sc[i] = colvalid ? s[i] : -INFINITY;

#pragma unroll
    for (int i = 0; i < 8; ++i) {
      const float tmax = row16_max(sc[i]);
      const float mnew = fmaxf(m_run[i], tmax);
      const float corr = __expf(m_run[i] - mnew);
      const float p    = __expf(sc[i] - mnew);
      const float tsum = row16_sum(p);
      l_run[i] = l_run[i] * corr + tsum;
      m_run[i] = mnew;
#pragma unroll
      for (int dc = 0; dc < 16; ++dc) o_acc[dc][i] *= corr;
      // stash P in natural [m][j] layout (row = i + 8*khalf, col = n)
      myps[(i + 8 * khalf) * PS_STRIDE + n] = p;
    }

    // ---- A operands for P (16x16, K-steps of 4), from per-wave LDS ----
    v2f ap[4];
#pragma unroll
    for (int kkp = 0; kkp < 4; ++kkp)
      ap[kkp] = *(const v2f*)(myps + n * PS_STRIDE + kkp * 4 + khalf * 2);

    // ---- O[16x256] += P(16x16) * V(16x256): 4 k-steps x 16 d-chunks ----
#pragma unroll
    for (int kkp = 0; kkp < 4; ++kkp) {
      int j0 = jt + kkp * 4 + khalf * 2;
      int ja = (j0     < STOT) ? j0     : STOT - 1;
      int jc = (j0 + 1 < STOT) ? j0 + 1 : STOT - 1;
      const float* vra = Vbase + (size_t)ja * HD;
      const float* vrb = Vbase + (size_t)jc * HD;
#pragma unroll
      for (int dc = 0; dc < 16; ++dc) {
        v2f b;
        b[0] = vra[dc * 16 + n];
        b[1] = vrb[dc * 16 + n];
        o_acc[dc] = wmma_f32_4(ap[kkp], b, o_acc[dc]);
      }
    }
  }

  // ---- write partial (m, l, O) — real rows 0..7 live in lanes 0..15 ----
  const size_t pidx = ((size_t)g * NCHUNK + chunk) * NWAVE + wave;
  float* pb0 = partials + pidx * PART_STRIDE;
  if (khalf == 0) {
#pragma unroll
    for (int i = 0; i < 8; ++i) {
      float* pb = pb0 + i * (2 + HD);
      if (lane == 0) { pb[0] = m_run[i]; pb[1] = l_run[i]; }
#pragma unroll
      for (int dc = 0; dc < 16; ++dc)
        pb[2 + dc * 16 + lane] = o_acc[dc][i];
    }
  }
}

// ---------------------------------------------------------------------------
// Kernel 4: combine partials. One block per q-head (16), one thread per d.
// ---------------------------------------------------------------------------
__global__ void attn_combine_kernel(const float* __restrict__ partials,
                                    float* __restrict__ attn) {
  const int h = blockIdx.x, d = threadIdx.x;
  const int g = h >> 3, r = h & 7;
  const float* pg = partials + (size_t)g * NPART_G * PART_STRIDE;

  float gmax = -INFINITY;
  for (int p = 0; p < NPART_G; ++p)
    gmax = fmaxf(gmax, pg[(size_t)p * PART_STRIDE + r * (2 + HD)]);

  float l = 0.0f, acc = 0.0f;
  for (int p = 0; p < NPART_G; ++p) {
    const float* pb = pg + (size_t)p * PART_STRIDE + r * (2 + HD);
    const float w = __expf(pb[0] - gmax);
    l   += pb[1] * w;
    acc += pb[2 + d] * w;
  }
  attn[h * HD + d] = acc / l;
}

// ---------------------------------------------------------------------------
// Kernel 5: output projection y = attn @ Wo. 8 blocks x 256 threads.
// ---------------------------------------------------------------------------
__global__ void out_proj_kernel(const float* __restrict__ attn,
                                const float* __restrict__ Wo,
                                float* __restrict__ y) {
  __shared__ float as[NHQ * HD];
  for (int i = threadIdx.x; i < NHQ * HD; i += 256) as[i] = attn[i];
  __syncthreads();

  const int j = blockIdx.x * 256 + threadIdx.x;
  float acc = 0.0f;
#pragma unroll 8
  for (int i = 0; i < NHQ * HD; ++i)
    acc += as[i] * __builtin_nontemporal_load(&Wo[(size_t)i * HID + j]);
  y[j] = acc;
}

// ---------------------------------------------------------------------------
extern "C" void kernel_launch(void* const* d_in, const int* in_sizes, int n_in,
                              void* d_out, int out_size, void* d_ws, size_t ws_size,
                              hipStream_t stream) {
  const float* x  = (const float*)d_in[0];
  const float* kc = (const float*)d_in[1];
  const float* vc = (const float*)d_in[2];
  const float* Wq = (const float*)d_in[3];
  const float* Wk = (const float*)d_in[4];
  const float* Wv = (const float*)d_in[5];
  const float* Wo = (const float*)d_in[6];

  float* y    = (float*)d_out;
  float* kout = y + HID;
  float* vout = kout + (long long)NKV * STOT * HD;
  float* ws   = (float*)d_ws;

  qkv_proj_kernel<<<20, 256, 0, stream>>>(x, Wq, Wk, Wv, ws);
  cache_copy_kernel<<<8192, 256, 0, stream>>>(kc, vc, ws + 4096, kout, vout);
  attn_partial_kernel<<<NKV * NCHUNK, 256, 0, stream>>>(ws, kout, vout, ws + WS_PART);
  attn_combine_kernel<<<NHQ, HD, 0, stream>>>(ws + WS_PART, ws + WS_ATTN);
  out_proj_kernel<<<HID / 256, 256, 0, stream>>>(ws + WS_ATTN, Wo, y);
}